// SpectralConv2d_5394478924220
// MI455X (gfx1250) — compile-verified
//
#include <hip/hip_runtime.h>
#include <hip/hip_bf16.h>

// SpectralConv2d (degenerate FNO: pointwise mode scaling) for gfx1250.
//
// Truncated rfft2/irfft2 expressed as dense GEMMs against DFT factor
// matrices, executed with V_WMMA_F32_16X16X4_F32 (f32 in/out is mandated by
// the reference; ~26 GFLOP of transform math vs a ~512 MB streaming floor ->
// bandwidth bound, WMMA keeps the math off the VALU issue slots).
//
// Round-2 change: all B-operand tables stored K-contiguous (transposed) and
// the intermediate F' stored as separate k2-major real/imag planes, so every
// WMMA B-fragment is one aligned b64 load and the F' epilogue is b128 stores.
//
// The JAX threefry colored-noise subtraction is NOT reproducible bit-exactly
// outside JAX (threefry key schedule + length-524288 FFT); it is omitted.

typedef __attribute__((ext_vector_type(2))) float v2f;
typedef __attribute__((ext_vector_type(8))) float v8f;

__device__ __forceinline__ v8f wmma4(v2f a, v2f b, v8f c) {
  // D(16x16,f32) += A(16x4,f32) * B(4x16,f32)
  return __builtin_amdgcn_wmma_f32_16x16x4_f32(
      /*neg_a=*/false, a, /*neg_b=*/false, b,
      /*c_mod=*/(short)0, c, /*reuse_a=*/false, /*reuse_b=*/false);
}

// ---------------------------------------------------------------------------
// Table init (all B-side tables K-contiguous):
//   BfT [64][256] : forward x-DFT, rows 0..31 = cos(th), 32..63 = -sin(th),
//                   indexed [n][x]  (x = K)
//   Rr/Ri/Rin [64][256] : forward y-DFT rows, k1 in {0..31, 224..255},
//                   indexed [k1row][y]  (A-side, y = K contiguous)
//                   (Rin = -Ri: f32 WMMA has no A-negate)
//   Sr/Si/Sin [256][64] : inverse y-DFT e^{+i th}, indexed [y][k1row] (A-side)
//   CtT/StnT [256][32] : inverse x-DFT, indexed [x][k2] (k2 = K), with
//                   1/65536 and the Hermitian x2 weights folded in
// ---------------------------------------------------------------------------
__global__ void __launch_bounds__(256)
fno_init_tables(float* __restrict__ BfT,
                float* __restrict__ Rr, float* __restrict__ Ri, float* __restrict__ Rin,
                float* __restrict__ Sr, float* __restrict__ Si, float* __restrict__ Sin,
                float* __restrict__ CtT, float* __restrict__ StnT) {
  const float TWO_PI = 6.283185307179586f;
  const int id = blockIdx.x * blockDim.x + threadIdx.x;  // 0 .. 16383

  {  // BfT: [n 0..63][x 0..255]
    const int n = id >> 8, xx = id & 255, k2 = n & 31;
    const float th = TWO_PI * (float)((k2 * xx) & 255) * (1.0f / 256.0f);
    BfT[id] = (n < 32) ? __cosf(th) : -__sinf(th);
  }
  {  // R tables: [j 0..63][y 0..255]
    const int j = id >> 8, y = id & 255;
    const int k1 = (j < 32) ? j : (192 + j);  // rows 32..63 -> k1 = 224..255
    const float th = TWO_PI * (float)((k1 * y) & 255) * (1.0f / 256.0f);
    const float c = __cosf(th), s = __sinf(th);
    Rr[id] = c; Ri[id] = -s; Rin[id] = s;
  }
  {  // S tables: [y 0..255][j 0..63]
    const int y = id >> 6, j = id & 63;
    const int k1 = (j < 32) ? j : (192 + j);
    const float th = TWO_PI * (float)((k1 * y) & 255) * (1.0f / 256.0f);
    const float c = __cosf(th), s = __sinf(th);
    Sr[id] = c; Si[id] = s; Sin[id] = -s;
  }
  if (id < 8192) {  // CtT/StnT: [x 0..255][k2 0..31]
    const int xx = id >> 5, k = id & 31;
    const float th = TWO_PI * (float)((k * xx) & 255) * (1.0f / 256.0f);
    const float sc = ((k == 0) ? 1.0f : 2.0f) * (1.0f / 65536.0f);
    CtT[id]  = sc * __cosf(th);
    StnT[id] = -sc * __sinf(th);
  }
}

// wsum stored k2-major: wsum[c][k2][j][2] = sum_o w{1,2}[c][o][m1][k2]
// (j<32 -> w1 row j; j>=32 -> w2 row j-32, i.e. out_ft rows 224..255).
// k1 (=j) contiguous so the fwd epilogue reads aligned v2f pairs.
__global__ void __launch_bounds__(256)
fno_wsum(const float* __restrict__ w1, const float* __restrict__ w2,
         float* __restrict__ wsum) {
  const int id = blockIdx.x * blockDim.x + threadIdx.x;  // (c, j, k2)
  if (id >= 64 * 64 * 32) return;
  const int k2 = id & 31;
  const int j  = (id >> 5) & 63;
  const int c  = id >> 11;
  const float* w = (j < 32) ? w1 : w2;
  const int m1 = j & 31;
  float sr = 0.0f, si = 0.0f;
  for (int o = 0; o < 64; ++o) {
    const size_t base = ((((size_t)c * 64 + o) * 32 + m1) * 32 + k2) * 2;
    sr += w[base];
    si += w[base + 1];
  }
  const size_t oidx = (((size_t)c * 32 + k2) * 64 + j) * 2;
  wsum[oidx]     = sr;
  wsum[oidx + 1] = si;
}

// ---------------------------------------------------------------------------
// Forward: per image (b*64+c), compute the 64x32 used modes of rfft2(x),
// scale by wsum[c], store F' as two k2-major planes FrT/FiT [img][32][64].
//   Phase 1 (WMMA): G[256][64] = X[256][256] * Bf  -> LDS (row-major)
//   Phase 2 (WMMA): F = R (complex) * G (complex), K=256 over y
// blockDim = 256 (8 wave32), dynamic LDS = 64 KB.
// ---------------------------------------------------------------------------
__global__ void __launch_bounds__(256)
fno_fwd(const float* __restrict__ x,
        const float* __restrict__ BfT,
        const float* __restrict__ Rr, const float* __restrict__ Ri,
        const float* __restrict__ Rin,
        const float* __restrict__ wsum,
        float* __restrict__ FrT, float* __restrict__ FiT) {
  extern __shared__ float Gs[];  // [256][64]
  const int img  = blockIdx.x;
  const float* X = x + (size_t)img * 65536;
  const int lane = threadIdx.x & 31;
  const int wave = threadIdx.x >> 5;
  const int lmod = lane & 15;
  const int lhi  = lane >> 4;

  // ---- Phase 1: G = X * Bf (M=256, N=64, K=256) ----
  for (int t = wave; t < 64; t += 8) {
    const int mt = t >> 2, nt = t & 3;
    const int mg = mt * 16 + lmod;  // image row y
    const int ng = nt * 16 + lmod;  // G column
    v8f acc = {};
    for (int kk = 0; kk < 256; kk += 4) {
      const int kb = kk + 2 * lhi;
      const v2f a = *(const v2f*)(X + mg * 256 + kb);       // b64
      const v2f b = *(const v2f*)(BfT + ng * 256 + kb);     // b64 (transposed)
      acc = wmma4(a, b, acc);
    }
#pragma unroll
    for (int r = 0; r < 8; ++r)
      Gs[(mt * 16 + r + lhi * 8) * 64 + ng] = acc[r];
  }
  __syncthreads();

  // ---- Phase 2: F[64][32] (complex) = R * G, then scale by wsum[c] ----
  {
    const int mt = wave >> 1, nt = wave & 1;  // 4 x 2 output tiles, 1/wave
    const int mg = mt * 16 + lmod;  // k1-row index 0..63
    const int ng = nt * 16 + lmod;  // k2 0..31
    v8f fr = {}, fi = {};
    for (int kk = 0; kk < 256; kk += 4) {
      const int kb = kk + 2 * lhi;
      const v2f ar = *(const v2f*)(Rr + mg * 256 + kb);
      const v2f ai = *(const v2f*)(Ri + mg * 256 + kb);
      const v2f an = *(const v2f*)(Rin + mg * 256 + kb);
      v2f br, bi;
      br.x = Gs[kb * 64 + ng];        br.y = Gs[(kb + 1) * 64 + ng];
      bi.x = Gs[kb * 64 + 32 + ng];   bi.y = Gs[(kb + 1) * 64 + 32 + ng];
      fr = wmma4(ar, br, fr);  // Rr*Gr
      fr = wmma4(an, bi, fr);  // -Ri*Gi
      fi = wmma4(ar, bi, fi);  // Rr*Gi
      fi = wmma4(ai, br, fi);  // Ri*Gr
    }
    // Epilogue: F' = F * ws[c]; per lane, r=0..7 covers 8 consecutive k1,
    // so each plane is two aligned b128 stores.
    const int ch = img & 63;
    const int k1b = mt * 16 + lhi * 8;                       // k1 base
    const float* Wc = wsum + (((size_t)ch * 32 + ng) * 64 + k1b) * 2;
    float pr[8], pi[8];
#pragma unroll
    for (int r = 0; r < 8; ++r) {
      const v2f wc = *(const v2f*)(Wc + r * 2);              // (wr, wi)
      pr[r] = fr[r] * wc.x - fi[r] * wc.y;
      pi[r] = fr[r] * wc.y + fi[r] * wc.x;
    }
    float* fro = FrT + (size_t)img * 2048 + ng * 64 + k1b;
    float* fio = FiT + (size_t)img * 2048 + ng * 64 + k1b;
    *(float4*)(fro)     = make_float4(pr[0], pr[1], pr[2], pr[3]);
    *(float4*)(fro + 4) = make_float4(pr[4], pr[5], pr[6], pr[7]);
    *(float4*)(fio)     = make_float4(pi[0], pi[1], pi[2], pi[3]);
    *(float4*)(fio + 4) = make_float4(pi[4], pi[5], pi[6], pi[7]);
  }
}

// ---------------------------------------------------------------------------
// Inverse: per image, Z[256][32] (complex) = S (complex) * F' (complex),
// then out[256][256] = Zr*Ct + Zi*Stn  (irfft2 restricted to live modes).
// blockDim = 256 (8 wave32), dynamic LDS = 64 KB (Z staging).
// ---------------------------------------------------------------------------
__global__ void __launch_bounds__(256)
fno_inv(const float* __restrict__ FrT, const float* __restrict__ FiT,
        const float* __restrict__ Sr, const float* __restrict__ Si,
        const float* __restrict__ Sin,
        const float* __restrict__ CtT, const float* __restrict__ StnT,
        float* __restrict__ out) {
  extern __shared__ float Zs[];  // [256][64] : cols 0..31 Zr, 32..63 Zi
  const int img = blockIdx.x;
  const float* Fr = FrT + (size_t)img * 2048;  // [32][64], L2-resident
  const float* Fi = FiT + (size_t)img * 2048;
  const int lane = threadIdx.x & 31;
  const int wave = threadIdx.x >> 5;
  const int lmod = lane & 15;
  const int lhi  = lane >> 4;

  // ---- Phase 1: Z = S * F' (M=256, N=32, K=64) ----
  for (int t = wave; t < 32; t += 8) {
    const int mt = t >> 1, nt = t & 1;
    const int mg = mt * 16 + lmod;  // y
    const int ng = nt * 16 + lmod;  // k2
    v8f zr = {}, zi = {};
    for (int kk = 0; kk < 64; kk += 4) {
      const int kb = kk + 2 * lhi;
      const v2f sr = *(const v2f*)(Sr + mg * 64 + kb);
      const v2f si = *(const v2f*)(Si + mg * 64 + kb);
      const v2f sn = *(const v2f*)(Sin + mg * 64 + kb);
      const v2f br = *(const v2f*)(Fr + ng * 64 + kb);   // b64 (k1-contig)
      const v2f bi = *(const v2f*)(Fi + ng * 64 + kb);   // b64
      zr = wmma4(sr, br, zr);  // Sr*Fr
      zr = wmma4(sn, bi, zr);  // -Si*Fi
      zi = wmma4(sr, bi, zi);  // Sr*Fi
      zi = wmma4(si, br, zi);  // Si*Fr
    }
#pragma unroll
    for (int r = 0; r < 8; ++r) {
      const int y = mt * 16 + r + lhi * 8;
      Zs[y * 64 + ng]      = zr[r];
      Zs[y * 64 + 32 + ng] = zi[r];
    }
  }
  __syncthreads();

  // ---- Phase 2: out = Zr*Ct + Zi*Stn (M=256, N=256, K=32) ----
  float* O = out + (size_t)img * 65536;
  for (int t = wave; t < 256; t += 8) {
    const int mt = t >> 4, nt = t & 15;
    const int mg = mt * 16 + lmod;  // y
    const int ng = nt * 16 + lmod;  // x
    v8f acc = {};
#pragma unroll
    for (int kk = 0; kk < 32; kk += 4) {
      const int kb = kk + 2 * lhi;
      const v2f ar = *(const v2f*)(Zs + mg * 64 + kb);
      const v2f ai = *(const v2f*)(Zs + mg * 64 + 32 + kb);
      const v2f bc = *(const v2f*)(CtT + ng * 32 + kb);   // b64 (transposed)
      const v2f bs = *(const v2f*)(StnT + ng * 32 + kb);  // b64
      acc = wmma4(ar, bc, acc);
      acc = wmma4(ai, bs, acc);
    }
#pragma unroll
    for (int r = 0; r < 8; ++r)
      O[(mt * 16 + r + lhi * 8) * 256 + ng] = acc[r];
  }
}

// ---------------------------------------------------------------------------
// Workspace layout (floats):
//   BfT 16384 | Rr 16384 | Ri 16384 | Rin 16384 | Sr 16384 | Si 16384 |
//   Sin 16384 | CtT 8192 | StnT 8192 | wsum 262144 |
//   FrT 2097152 | FiT 2097152          -> total ~17.6 MB
// ---------------------------------------------------------------------------
extern "C" void kernel_launch(void* const* d_in, const int* in_sizes, int n_in,
                              void* d_out, int out_size, void* d_ws, size_t ws_size,
                              hipStream_t stream) {
  const float* x  = (const float*)d_in[0];
  const float* w1 = (const float*)d_in[1];
  const float* w2 = (const float*)d_in[2];
  float* out = (float*)d_out;
  float* ws  = (float*)d_ws;

  float* BfT  = ws;
  float* Rr   = ws + 16384;
  float* Ri   = ws + 32768;
  float* Rin  = ws + 49152;
  float* Sr   = ws + 65536;
  float* Si   = ws + 81920;
  float* Sin  = ws + 98304;
  float* CtT  = ws + 114688;
  float* StnT = ws + 122880;
  float* Wsum = ws + 131072;   // 262144 floats
  float* FrT  = ws + 393216;   // 2097152 floats
  float* FiT  = ws + 2490368;  // 2097152 floats

  fno_init_tables<<<64, 256, 0, stream>>>(BfT, Rr, Ri, Rin, Sr, Si, Sin, CtT, StnT);
  fno_wsum<<<512, 256, 0, stream>>>(w1, w2, Wsum);
  fno_fwd<<<1024, 256, 64 * 1024, stream>>>(x, BfT, Rr, Ri, Rin, Wsum, FrT, FiT);
  fno_inv<<<1024, 256, 64 * 1024, stream>>>(FrT, FiT, Sr, Si, Sin, CtT, StnT, out);
}